// HypergraphFusionModel_89421219103603
// MI455X (gfx1250) — compile-verified
//
#include <hip/hip_runtime.h>

typedef float v2f __attribute__((ext_vector_type(2)));
typedef float v4f __attribute__((ext_vector_type(4)));
typedef float v8f __attribute__((ext_vector_type(8)));

constexpr int NB   = 16;    // batch
constexpr int CC   = 1024;  // nodes / hyperedges
constexpr int DIN  = 512;
constexpr int DH   = 1024;
constexpr int DOUT = 512;

// ---------------------------------------------------------------------------
// Stage 1a: partial column-sums of x over c.  s1p[chunk][n][d], 8 chunks of 128 c.
// ---------------------------------------------------------------------------
__global__ void k_partial_sum_x(const float* __restrict__ x, float* __restrict__ s1p) {
    const int n     = blockIdx.x & 15;
    const int chunk = blockIdx.x >> 4;       // 0..7
    const int d     = threadIdx.x;           // 0..511
    const float* px = x + (size_t)n * CC * DIN + (size_t)chunk * 128 * DIN + d;
    float s = 0.f;
    #pragma unroll 8
    for (int c = 0; c < 128; ++c) s += px[(size_t)c * DIN];
    s1p[((size_t)chunk * NB + n) * DIN + d] = s;
}

// Stage 1b: reduce the 8 partials -> s1[n][d]  ([16,512])
__global__ void k_reduce_s1(const float* __restrict__ s1p, float* __restrict__ s1) {
    const int i = blockIdx.x * blockDim.x + threadIdx.x;   // 0..8191  (n*512+d)
    float s = 0.f;
    #pragma unroll
    for (int ch = 0; ch < 8; ++ch) s += s1p[(size_t)ch * (NB * DIN) + i];
    s1[i] = s;
}

// Stage 1c: rSw[n] = 1 / sum_e adj[n,0,e]  (one wave per n, shuffle reduction).
// Reference computes the reciprocal degree once and multiplies; do the same.
__global__ void k_sw(const float* __restrict__ adj, float* __restrict__ rSw) {
    const int n    = blockIdx.x;
    const int lane = threadIdx.x;            // 0..31
    const float* pa = adj + (size_t)n * CC * CC;   // row 0 of adj[n]
    float s = 0.f;
    for (int e = lane; e < CC; e += 32) s += pa[e];
    #pragma unroll
    for (int off = 16; off > 0; off >>= 1) s += __shfl_down(s, off, 32);
    if (lane == 0) rSw[n] = (s > 0.f) ? (1.0f / s) : 0.f;
}

// ---------------------------------------------------------------------------
// Stage 2: u = s1 @ W1^T, scale by rSw, +b1, relu  -> r1 [16,1024]
// One wave per 16-column tile; M=16 == batch fits one WMMA tile.
// V_WMMA_F32_16X16X4_F32, K swept 4 at a time (full fp32 precision).
// ---------------------------------------------------------------------------
__global__ void k_gemm1_relu(const float* __restrict__ s1, const float* __restrict__ W1,
                             const float* __restrict__ b1, const float* __restrict__ rSw,
                             float* __restrict__ r1) {
    const int lane = threadIdx.x;            // 0..31, wave fully active (EXEC all 1s)
    const int tile = blockIdx.x;             // 0..63
    const int low  = lane & 15;
    const int hi   = lane >> 4;
    const int col  = tile * 16 + low;        // output column (h)
    const float* pa = s1 + (size_t)low * DIN;        // A row m = lane&15
    const float* pb = W1 + (size_t)col * DIN;        // B(k,col) = W1[col][k]
    v8f acc = {};
    #pragma unroll 4
    for (int kk = 0; kk < DIN; kk += 4) {
        const int k0 = kk + 2 * hi;                  // VGPR j holds K = 2*hi + j
        v2f a = { pa[k0], pa[k0 + 1] };
        v2f b = { pb[k0], pb[k0 + 1] };
        acc = __builtin_amdgcn_wmma_f32_16x16x4_f32(false, a, false, b,
                                                    (short)0, acc, false, false);
    }
    const float bias = b1[col];
    #pragma unroll
    for (int r = 0; r < 8; ++r) {
        const int Mrow = r + 8 * hi;                 // C/D: M = r + 8*(lane>=16)
        float u = acc[r] * rSw[Mrow] + bias;
        u = u > 0.f ? u : 0.f;                       // relu
        r1[(size_t)Mrow * DH + col] = u;
    }
}

// ---------------------------------------------------------------------------
// Stage 3: LayerNorm over DH=1024 per batch row (rows identical across c).
// ---------------------------------------------------------------------------
__global__ void k_ln(const float* __restrict__ r1, const float* __restrict__ g,
                     const float* __restrict__ bb, float* __restrict__ lnout) {
    __shared__ float ssum[256];
    __shared__ float ssq[256];
    const int n = blockIdx.x, t = threadIdx.x;
    const float* row = r1 + (size_t)n * DH;
    float vals[4];
    float s = 0.f, q = 0.f;
    #pragma unroll
    for (int i = 0; i < 4; ++i) {
        float xv = row[t + i * 256];
        vals[i] = xv; s += xv; q += xv * xv;
    }
    ssum[t] = s; ssq[t] = q; __syncthreads();
    for (int off = 128; off > 0; off >>= 1) {
        if (t < off) { ssum[t] += ssum[t + off]; ssq[t] += ssq[t + off]; }
        __syncthreads();
    }
    const float mean = ssum[0] * (1.0f / DH);
    const float var  = ssq[0] * (1.0f / DH) - mean * mean;
    const float inv  = rsqrtf(var + 1e-5f);
    #pragma unroll
    for (int i = 0; i < 4; ++i) {
        const int h = t + i * 256;
        lnout[(size_t)n * DH + h] = (vals[i] - mean) * inv * g[h] + bb[h];
    }
}

// ---------------------------------------------------------------------------
// Stage 4: v = (C * rSw[n]) * (ln @ W2^T) + b2   -> v [16,512]
// ---------------------------------------------------------------------------
__global__ void k_gemm2(const float* __restrict__ ln, const float* __restrict__ W2,
                        const float* __restrict__ b2, const float* __restrict__ rSw,
                        float* __restrict__ v) {
    const int lane = threadIdx.x;
    const int tile = blockIdx.x;             // 0..31
    const int low  = lane & 15;
    const int hi   = lane >> 4;
    const int col  = tile * 16 + low;
    const float* pa = ln + (size_t)low * DH;
    const float* pb = W2 + (size_t)col * DH;         // W2 is [DOUT, DH] row-major
    v8f acc = {};
    #pragma unroll 4
    for (int kk = 0; kk < DH; kk += 4) {
        const int k0 = kk + 2 * hi;
        v2f a = { pa[k0], pa[k0 + 1] };
        v2f b = { pb[k0], pb[k0 + 1] };
        acc = __builtin_amdgcn_wmma_f32_16x16x4_f32(false, a, false, b,
                                                    (short)0, acc, false, false);
    }
    const float bias = b2[col];
    #pragma unroll
    for (int r = 0; r < 8; ++r) {
        const int Mrow = r + 8 * hi;
        v[(size_t)Mrow * DOUT + col] = acc[r] * ((float)CC * rSw[Mrow]) + bias;
    }
}

// ---------------------------------------------------------------------------
// Stage 5: out[n,c,d] = x[n,c,d] + v[n,d]   (67 MB stream -> NT hints)
// ---------------------------------------------------------------------------
__global__ void k_final(const float* __restrict__ x, const float* __restrict__ v,
                        float* __restrict__ out) {
    const size_t i4 = (size_t)blockIdx.x * blockDim.x + threadIdx.x; // float4 index
    const int    d4 = (int)(i4 & 127);       // DIN/4 = 128
    const size_t nc = i4 >> 7;
    const int    n  = (int)(nc >> 10);       // / C
    const v4f* xv = (const v4f*)x;
    const v4f* vv = (const v4f*)v;
    v4f xx = __builtin_nontemporal_load(&xv[i4]);
    v4f av = vv[(size_t)n * 128 + d4];
    __builtin_nontemporal_store(xx + av, ((v4f*)out) + i4);
}

// ---------------------------------------------------------------------------
extern "C" void kernel_launch(void* const* d_in, const int* in_sizes, int n_in,
                              void* d_out, int out_size, void* d_ws, size_t ws_size,
                              hipStream_t stream) {
    const float* x     = (const float*)d_in[0];
    const float* adj   = (const float*)d_in[1];
    const float* W1    = (const float*)d_in[2];
    const float* b1    = (const float*)d_in[3];
    const float* gamma = (const float*)d_in[4];
    const float* beta  = (const float*)d_in[5];
    const float* W2    = (const float*)d_in[6];
    const float* b2    = (const float*)d_in[7];
    float* out = (float*)d_out;

    float* ws  = (float*)d_ws;
    float* s1p = ws;                         // 8*16*512  = 65536
    float* s1  = s1p + 8 * NB * DIN;         // 16*512    = 8192
    float* rSw = s1 + NB * DIN;              // 16 (pad 32)
    float* r1  = rSw + 32;                   // 16*1024
    float* lnb = r1 + NB * DH;               // 16*1024
    float* vb  = lnb + NB * DH;              // 16*512

    k_partial_sum_x<<<dim3(8 * NB), dim3(DIN), 0, stream>>>(x, s1p);
    k_reduce_s1   <<<dim3(NB),      dim3(DIN), 0, stream>>>(s1p, s1);
    k_sw          <<<dim3(NB),      dim3(32),  0, stream>>>(adj, rSw);
    k_gemm1_relu  <<<dim3(DH / 16), dim3(32),  0, stream>>>(s1, W1, b1, rSw, r1);
    k_ln          <<<dim3(NB),      dim3(256), 0, stream>>>(r1, gamma, beta, lnb);
    k_gemm2       <<<dim3(DOUT/16), dim3(32),  0, stream>>>(lnb, W2, b2, rSw, vb);
    k_final       <<<dim3((NB * CC * DIN / 4) / 256), dim3(256), 0, stream>>>(x, vb, out);
}